// DoubleConv_67800353734680
// MI455X (gfx1250) — compile-verified
//
#include <hip/hip_runtime.h>
#include <hip/hip_bf16.h>
#include <stddef.h>

// Problem constants (match reference setup_inputs).
#define BATCH 64
#define C1    256     // conv1 input channels
#define O1    512     // conv1 output channels
#define KW    5       // conv1 kernel width
#define LEN   2048    // sequence length
#define C2    1024    // conv2 input channels (2*O1)

// Tiling.
#define NTILE  128                 // l-columns per workgroup / per wave
#define NT2    (NTILE / 16)        // 8 WMMA n-tiles per wave
#define CHUNK  16                  // channels staged in LDS per iteration
#define TW     (NTILE + KW - 1)    // 132: staged width covers all 5 taps
#define XPITCH 144                 // float2 row pitch; 288 dwords = 32 mod 64
                                   // -> the two half-wave bank windows are disjoint

typedef float v2f __attribute__((ext_vector_type(2)));
typedef float v8f __attribute__((ext_vector_type(8)));

// ---------------------------------------------------------------------------
// Weight repack kernels (run once per launch; deterministic).
// W1r[(k*(C1/2)+cp)*O1 + o] = { W1[o][2cp][k], W1[o][2cp+1][k] }   (v2f)
// W2r[cp*O1 + o]            = { W2[o][2cp],    W2[o][2cp+1]    }   (v2f)
// This makes every WMMA A fragment a single aligned, coalesced b64 load.
// ---------------------------------------------------------------------------
__global__ __launch_bounds__(256) void repack_w1(const float* __restrict__ W1,
                                                 float* __restrict__ W1r) {
  const int t = blockIdx.x * 256 + threadIdx.x;          // KW*(C1/2)*O1 total
  if (t >= KW * (C1 / 2) * O1) return;
  const int o    = t % O1;
  const int rest = t / O1;
  const int cp   = rest % (C1 / 2);
  const int k    = rest / (C1 / 2);
  v2f v;
  v.x = W1[(size_t)o * (C1 * KW) + (size_t)(2 * cp) * KW + k];
  v.y = W1[(size_t)o * (C1 * KW) + (size_t)(2 * cp + 1) * KW + k];
  ((v2f*)W1r)[t] = v;
}

__global__ __launch_bounds__(256) void repack_w2(const float* __restrict__ W2,
                                                 float* __restrict__ W2r) {
  const int t = blockIdx.x * 256 + threadIdx.x;          // (C2/2)*O1 total
  if (t >= (C2 / 2) * O1) return;
  const int o  = t % O1;
  const int cp = t / O1;
  v2f v;
  v.x = W2[(size_t)o * C2 + 2 * cp];
  v.y = W2[(size_t)o * C2 + 2 * cp + 1];
  ((v2f*)W2r)[t] = v;
}

// ---------------------------------------------------------------------------
// Kernel 1: conv1d (K=5, pad=2) as implicit GEMM on V_WMMA_F32_16X16X4_F32.
// grid = (LEN/128, O1/64, BATCH), block = 128 (4 waves). Wave w computes the
// 16x128 tile rows [o0+16w, o0+16w+16) x cols [l0, l0+128).
// LDS holds x pair-interleaved over channels: one ds_load_b64 per B fragment.
// ---------------------------------------------------------------------------
__global__ __launch_bounds__(128) void conv1_wmma(const float* __restrict__ x,
                                                  const float* __restrict__ w1r,
                                                  float* __restrict__ h) {
  __shared__ v2f xs[CHUNK / 2][XPITCH];   // 8 x 144 float2 = 9216 B
  float* xsf = (float*)xs;

  const int b    = blockIdx.z;
  const int l0   = blockIdx.x * NTILE;
  const int o0   = blockIdx.y * 64;
  const int tid  = threadIdx.x;
  const int wave = tid >> 5;
  const int lane = tid & 31;
  const int ln15 = lane & 15;
  const int half = lane >> 4;                 // 0: K rows {0,1}; 1: K rows {2,3}
  const int ow   = o0 + wave * 16 + ln15;     // A-fragment row (M = lane&15)

  const float* xb  = x + (size_t)b * C1 * LEN;
  const v2f*   w1v = (const v2f*)w1r;

  const v8f vzero = {0.f,0.f,0.f,0.f,0.f,0.f,0.f,0.f};
  v8f acc[NT2];
  #pragma unroll
  for (int nt = 0; nt < NT2; ++nt) acc[nt] = vzero;

  for (int c0 = 0; c0 < C1; c0 += CHUNK) {
    // Stage x[c0..c0+15][l0-2 .. l0+129] (zero-padded), pair-interleaved.
    for (int t = tid; t < CHUNK * TW; t += 128) {
      const int cc = t / TW;
      const int j  = t - cc * TW;
      const int gl = l0 - (KW / 2) + j;
      float v = 0.f;
      if (gl >= 0 && gl < LEN) v = xb[(size_t)(c0 + cc) * LEN + gl];
      xsf[(size_t)(cc >> 1) * (2 * XPITCH) + (j << 1) + (cc & 1)] = v;
    }
    __syncthreads();

    for (int k = 0; k < KW; ++k) {
      #pragma unroll
      for (int ccp = 0; ccp < CHUNK / 4; ++ccp) {      // 4 K-steps of 4 channels
        const int cpA = (c0 >> 1) + ccp * 2 + half;
        const v2f a = w1v[(size_t)(k * (C1 / 2) + cpA) * O1 + ow];
        const int cpB = ccp * 2 + half;
        #pragma unroll
        for (int nt = 0; nt < NT2; ++nt) {
          const v2f bf = xs[cpB][nt * 16 + ln15 + k];  // single ds_load_b64
          acc[nt] = __builtin_amdgcn_wmma_f32_16x16x4_f32(
              false, a, false, bf, (short)0, acc[nt], false, false);
        }
      }
    }
    __syncthreads();
  }

  // C/D layout: VGPR j holds M = j + 8*(lane>=16), N = lane&15.
  const int mhi = half * 8;
  float* hb = h + (size_t)b * O1 * LEN;
  #pragma unroll
  for (int nt = 0; nt < NT2; ++nt) {
    const int col = l0 + nt * 16 + ln15;
    #pragma unroll
    for (int j = 0; j < 8; ++j) {
      hb[(size_t)(o0 + wave * 16 + mhi + j) * LEN + col] = acc[nt][j];
    }
  }
}

// ---------------------------------------------------------------------------
// Kernel 2: in-place InstanceNorm (biased var, eps=1e-5) + ReLU per (b,o) row.
// One 256-thread block per row; b128 loads/stores; values cached in registers.
// ---------------------------------------------------------------------------
__global__ __launch_bounds__(256) void inorm_relu(float* __restrict__ y) {
  float4* y4 = (float4*)(y + (size_t)blockIdx.x * LEN);
  const int tid = threadIdx.x;

  const float4 a = y4[tid];
  const float4 c = y4[tid + 256];
  float s = a.x + a.y + a.z + a.w + c.x + c.y + c.z + c.w;
  float q = a.x*a.x + a.y*a.y + a.z*a.z + a.w*a.w
          + c.x*c.x + c.y*c.y + c.z*c.z + c.w*c.w;

  __shared__ float rs[256], rq[256];
  rs[tid] = s;
  rq[tid] = q;
  __syncthreads();
  for (int off = 128; off > 0; off >>= 1) {
    if (tid < off) {
      rs[tid] += rs[tid + off];
      rq[tid] += rq[tid + off];
    }
    __syncthreads();
  }

  const float mean = rs[0] * (1.f / LEN);
  const float var  = rq[0] * (1.f / LEN) - mean * mean;
  const float inv  = rsqrtf(var + 1e-5f);

  float4 oa, oc;
  oa.x = fmaxf((a.x - mean) * inv, 0.f); oa.y = fmaxf((a.y - mean) * inv, 0.f);
  oa.z = fmaxf((a.z - mean) * inv, 0.f); oa.w = fmaxf((a.w - mean) * inv, 0.f);
  oc.x = fmaxf((c.x - mean) * inv, 0.f); oc.y = fmaxf((c.y - mean) * inv, 0.f);
  oc.z = fmaxf((c.z - mean) * inv, 0.f); oc.w = fmaxf((c.w - mean) * inv, 0.f);
  y4[tid]       = oa;
  y4[tid + 256] = oc;
}

// ---------------------------------------------------------------------------
// Kernel 3: 1x1 conv over concat([h, gather(h, idx)]) as WMMA GEMM, K=1024.
// For chunks with c >= 512 the LDS staging performs the neighbour gather, so
// the concatenated tensor is never materialized.
// ---------------------------------------------------------------------------
__global__ __launch_bounds__(128) void conv2_wmma(const float* __restrict__ h,
                                                  const int* __restrict__ idx,
                                                  const float* __restrict__ w2r,
                                                  float* __restrict__ out) {
  __shared__ v2f hs[CHUNK / 2][XPITCH];
  __shared__ int sidx[NTILE];
  float* hsf = (float*)hs;

  const int b    = blockIdx.z;
  const int l0   = blockIdx.x * NTILE;
  const int o0   = blockIdx.y * 64;
  const int tid  = threadIdx.x;
  const int wave = tid >> 5;
  const int lane = tid & 31;
  const int ln15 = lane & 15;
  const int half = lane >> 4;
  const int ow   = o0 + wave * 16 + ln15;

  const float* hb  = h + (size_t)b * O1 * LEN;
  const v2f*   w2v = (const v2f*)w2r;

  if (tid < NTILE) {
    const int iv = idx[(size_t)b * LEN + l0 + tid];
    sidx[tid] = ((unsigned)iv < (unsigned)LEN) ? iv : -1;  // idx==L -> padded 0
  }
  __syncthreads();

  const v8f vzero = {0.f,0.f,0.f,0.f,0.f,0.f,0.f,0.f};
  v8f acc[NT2];
  #pragma unroll
  for (int nt = 0; nt < NT2; ++nt) acc[nt] = vzero;

  for (int c0 = 0; c0 < C2; c0 += CHUNK) {
    const bool gather = (c0 >= O1);
    const int  crow   = gather ? (c0 - O1) : c0;

    for (int t = tid; t < CHUNK * NTILE; t += 128) {   // exactly 16 per thread
      const int cc = t >> 7;      // / 128
      const int j  = t & 127;     // % 128
      float v;
      if (!gather) {
        v = hb[(size_t)(crow + cc) * LEN + (l0 + j)];
      } else {
        const int iv = sidx[j];
        v = (iv >= 0) ? hb[(size_t)(crow + cc) * LEN + iv] : 0.f;
      }
      hsf[(size_t)(cc >> 1) * (2 * XPITCH) + (j << 1) + (cc & 1)] = v;
    }
    __syncthreads();

    #pragma unroll
    for (int ccp = 0; ccp < CHUNK / 4; ++ccp) {
      const int cpA = (c0 >> 1) + ccp * 2 + half;
      const v2f a = w2v[(size_t)cpA * O1 + ow];
      const int cpB = ccp * 2 + half;
      #pragma unroll
      for (int nt = 0; nt < NT2; ++nt) {
        const v2f bf = hs[cpB][nt * 16 + ln15];
        acc[nt] = __builtin_amdgcn_wmma_f32_16x16x4_f32(
            false, a, false, bf, (short)0, acc[nt], false, false);
      }
    }
    __syncthreads();
  }

  const int mhi = half * 8;
  float* ob = out + (size_t)b * O1 * LEN;
  #pragma unroll
  for (int nt = 0; nt < NT2; ++nt) {
    const int col = l0 + nt * 16 + ln15;
    #pragma unroll
    for (int j = 0; j < 8; ++j) {
      ob[(size_t)(o0 + wave * 16 + mhi + j) * LEN + col] = acc[nt][j];
    }
  }
}

// ---------------------------------------------------------------------------
// Launch: repack weights -> conv1 -> IN+ReLU (ws) -> conv2(+gather) -> IN+ReLU.
// Biases b1/b2 are mathematically cancelled by InstanceNorm and are skipped.
// Workspace layout (floats):
//   [0, 64*512*2048)                       h_raw           (256 MiB)
//   [H_END, +KW*(C1/2)*O1*2)               W1r pairs       (2.6 MiB)
//   [W1R_END, +(C2/2)*O1*2)                W2r pairs       (2.0 MiB)
// ---------------------------------------------------------------------------
extern "C" void kernel_launch(void* const* d_in, const int* in_sizes, int n_in,
                              void* d_out, int out_size, void* d_ws, size_t ws_size,
                              hipStream_t stream) {
  const float* x   = (const float*)d_in[0];
  const int*   idx = (const int*)d_in[1];
  const float* W1  = (const float*)d_in[2];
  // d_in[3] = b1 (zeros; no effect under InstanceNorm)
  const float* W2  = (const float*)d_in[4];
  // d_in[5] = b2 (zeros; no effect under InstanceNorm)
  float* out  = (float*)d_out;
  float* ws   = (float*)d_ws;

  const size_t H_FLOATS   = (size_t)BATCH * O1 * LEN;          // 67,108,864
  const size_t W1R_FLOATS = (size_t)KW * (C1 / 2) * O1 * 2;    // 655,360
  float* hbuf = ws;
  float* w1r  = ws + H_FLOATS;
  float* w2r  = w1r + W1R_FLOATS;

  repack_w1<<<(KW * (C1 / 2) * O1 + 255) / 256, 256, 0, stream>>>(W1, w1r);
  repack_w2<<<((C2 / 2) * O1 + 255) / 256, 256, 0, stream>>>(W2, w2r);

  dim3 grid(LEN / NTILE, O1 / 64, BATCH);  // (16, 8, 64)

  conv1_wmma<<<grid, 128, 0, stream>>>(x, w1r, hbuf);
  inorm_relu<<<BATCH * O1, 256, 0, stream>>>(hbuf);
  conv2_wmma<<<grid, 128, 0, stream>>>(hbuf, idx, w2r, out);
  inorm_relu<<<BATCH * O1, 256, 0, stream>>>(out);
}